// GraphCELoss_22239340658746
// MI455X (gfx1250) — compile-verified
//
#include <hip/hip_runtime.h>
#include <cstdint>

#define NSEG 262144
#define TILE 2048   // elements per tile (256 threads x 8)
#define TPB  8      // tiles per block in the streaming pass

// ---- monotone float<->uint encoding so float max can use global_atomic_max_u32 ----
__device__ __forceinline__ unsigned fenc(float f) {
  unsigned u = __float_as_uint(f);
  return (u & 0x80000000u) ? ~u : (u | 0x80000000u);
}
__device__ __forceinline__ float fdec(unsigned u) {
  unsigned b = (u & 0x80000000u) ? (u & 0x7fffffffu) : ~u;
  return __uint_as_float(b);
}

// ---- CDNA5 async global->LDS copy (tracked by ASYNCcnt, not LOADcnt) ----
__device__ __forceinline__ unsigned lds_off(const void* p) {
  // flat LDS-aperture addresses carry the LDS byte offset in the low 32 bits
  return (unsigned)(uintptr_t)p;
}
__device__ __forceinline__ void async_b128(unsigned lds_addr, const void* gaddr) {
  asm volatile("global_load_async_to_lds_b128 %0, %1, off"
               :: "v"(lds_addr), "v"((unsigned long long)(uintptr_t)gaddr)
               : "memory");
}

// ---------------- K0: init per-segment state + output ----------------
__global__ __launch_bounds__(256) void k_init(unsigned* __restrict__ mkey,
                                              float* __restrict__ S,
                                              float* __restrict__ T,
                                              float* __restrict__ U,
                                              float* __restrict__ out) {
  int i = blockIdx.x * 256 + threadIdx.x;
  if (i < NSEG) { mkey[i] = 0u; S[i] = 0.f; T[i] = 0.f; U[i] = 0.f; }
  if (i == 0) out[0] = 0.f;
}

// ---------------- K1: segmented max (run-compressed atomics) ----------------
__global__ __launch_bounds__(256) void k_segmax(const float* __restrict__ src,
                                                const int* __restrict__ idx,
                                                unsigned* __restrict__ mkey) {
  const int base = (blockIdx.x * 256 + threadIdx.x) * 8;
  const float4 a = *(const float4*)(src + base);
  const float4 b = *(const float4*)(src + base + 4);
  const int4 ia = *(const int4*)(idx + base);
  const int4 ib = *(const int4*)(idx + base + 4);
  float v[8] = {a.x, a.y, a.z, a.w, b.x, b.y, b.z, b.w};
  int  ix[8] = {ia.x, ia.y, ia.z, ia.w, ib.x, ib.y, ib.z, ib.w};
  int cur = ix[0];
  float m = v[0];
#pragma unroll
  for (int k = 1; k < 8; ++k) {
    if (ix[k] == cur) {
      m = fmaxf(m, v[k]);
    } else {
      atomicMax(&mkey[cur], fenc(m));
      cur = ix[k]; m = v[k];
    }
  }
  atomicMax(&mkey[cur], fenc(m));
}

// ---------------- K2: streaming pass with async double-buffered LDS staging ----------------
__global__ __launch_bounds__(256) void k_accum(const float* __restrict__ src,
                                               const float* __restrict__ tgt,
                                               const int* __restrict__ idx,
                                               const unsigned* __restrict__ mkey,
                                               float* __restrict__ S,
                                               float* __restrict__ T,
                                               float* __restrict__ U) {
  __shared__ float ls[2][TILE];
  __shared__ float lt[2][TILE];
  __shared__ int   li[2][TILE];
  const int o = threadIdx.x * 8;
  const size_t blk0 = (size_t)blockIdx.x * TPB * TILE;

  auto stage = [&](int buf, size_t ebase) {
    async_b128(lds_off(&ls[buf][o]),     src + ebase + o);
    async_b128(lds_off(&ls[buf][o + 4]), src + ebase + o + 4);
    async_b128(lds_off(&lt[buf][o]),     tgt + ebase + o);
    async_b128(lds_off(&lt[buf][o + 4]), tgt + ebase + o + 4);
    async_b128(lds_off(&li[buf][o]),     idx + ebase + o);
    async_b128(lds_off(&li[buf][o + 4]), idx + ebase + o + 4);
  };

  stage(0, blk0);
  int buf = 0;
  for (int t = 0; t < TPB; ++t) {
    if (t + 1 < TPB) {
      stage(buf ^ 1, blk0 + (size_t)(t + 1) * TILE);
      // 6 ops just issued for next tile; wait for everything older (this tile)
      asm volatile("s_wait_asynccnt 0x6" ::: "memory");
    } else {
      asm volatile("s_wait_asynccnt 0x0" ::: "memory");
    }
    // each wave consumes only the slice it staged -> no block barrier needed
    const float4 a  = *(const float4*)&ls[buf][o];
    const float4 b  = *(const float4*)&ls[buf][o + 4];
    const float4 wa = *(const float4*)&lt[buf][o];
    const float4 wb = *(const float4*)&lt[buf][o + 4];
    const int4  ia  = *(const int4*)&li[buf][o];
    const int4  ib  = *(const int4*)&li[buf][o + 4];
    float v[8] = {a.x, a.y, a.z, a.w, b.x, b.y, b.z, b.w};
    float w[8] = {wa.x, wa.y, wa.z, wa.w, wb.x, wb.y, wb.z, wb.w};
    int  ix[8] = {ia.x, ia.y, ia.z, ia.w, ib.x, ib.y, ib.z, ib.w};

    int cur = ix[0];
    float m = fdec(mkey[cur]);
    float d = v[0] - m;
    float sl = __expf(d), tl = w[0], ul = w[0] * d;
#pragma unroll
    for (int k = 1; k < 8; ++k) {
      if (ix[k] == cur) {
        d = v[k] - m;
        sl += __expf(d); tl += w[k]; ul += w[k] * d;
      } else {
        unsafeAtomicAdd(&S[cur], sl);
        unsafeAtomicAdd(&T[cur], tl);
        unsafeAtomicAdd(&U[cur], ul);
        cur = ix[k];
        m = fdec(mkey[cur]);
        d = v[k] - m;
        sl = __expf(d); tl = w[k]; ul = w[k] * d;
      }
    }
    unsafeAtomicAdd(&S[cur], sl);
    unsafeAtomicAdd(&T[cur], tl);
    unsafeAtomicAdd(&U[cur], ul);
    buf ^= 1;
  }
}

// ---------------- K3: per-segment loss + mean reduction ----------------
__global__ __launch_bounds__(256) void k_final(const float* __restrict__ S,
                                               const float* __restrict__ T,
                                               const float* __restrict__ U,
                                               float* __restrict__ out) {
  const int G = gridDim.x * 256;
  int i = blockIdx.x * 256 + threadIdx.x;
  float acc = 0.f;
  for (int j = i; j < NSEG; j += G) {
    float s = S[j];
    if (s > 0.f) acc += T[j] * __logf(s) - U[j];  // = per_seg contribution, scaled by 1/NSEG below
  }
  // wave32 reduction
#pragma unroll
  for (int off = 16; off > 0; off >>= 1)
    acc += __shfl_down(acc, off, 32);
  if ((threadIdx.x & 31) == 0)
    unsafeAtomicAdd(out, acc * (1.0f / NSEG));
}

extern "C" void kernel_launch(void* const* d_in, const int* in_sizes, int n_in,
                              void* d_out, int out_size, void* d_ws, size_t ws_size,
                              hipStream_t stream) {
  (void)n_in; (void)out_size; (void)ws_size;
  const float* src = (const float*)d_in[0];
  const int*   idx = (const int*)d_in[1];
  const float* tgt = (const float*)d_in[2];
  float* out = (float*)d_out;
  const int E = in_sizes[0];  // 16777216, multiple of TILE*TPB

  unsigned* mkey = (unsigned*)d_ws;
  float* S = (float*)((char*)d_ws + (size_t)NSEG * 4);
  float* T = (float*)((char*)d_ws + (size_t)NSEG * 8);
  float* U = (float*)((char*)d_ws + (size_t)NSEG * 12);

  k_init  <<<NSEG / 256, 256, 0, stream>>>(mkey, S, T, U, out);
  k_segmax<<<E / 2048, 256, 0, stream>>>(src, idx, mkey);
  k_accum <<<E / (TILE * TPB), 256, 0, stream>>>(src, tgt, idx, mkey, S, T, U);
  k_final <<<256, 256, 0, stream>>>(S, T, U, out);
}